// CRMF_9165460210142
// MI455X (gfx1250) — compile-verified
//
#include <hip/hip_runtime.h>

#define BSZ  16
#define NH   4
#define NN   1024
#define FIN  32
#define FOUT 64
#define NEG  0.2f

typedef __attribute__((ext_vector_type(2))) float v2f;
typedef __attribute__((ext_vector_type(8))) float v8f;

// h_prime is kept in "pair-interleaved" layout per (b,h):
//   hpG[pair][col][par]  with pair = j>>1, par = j&1, col = 0..63
// so the WMMA-B fragment (hp[j][c], hp[j+1][c]) is 8 contiguous bytes.

// low 32 bits of a flat LDS pointer == wave-relative LDS byte offset (ISA 10.2)
__device__ __forceinline__ unsigned ldsOffset(const void* p) {
  return (unsigned)(unsigned long long)(uintptr_t)p;
}

// Issue 16 async b128 copies: 32 KB contiguous global -> LDS (per 128-thr block).
// Same 24-bit inst offset is added to both the global and LDS address (ISA 15.18.3).
__device__ __forceinline__ void async_stage32k(const float* g, unsigned ldsB, int tid) {
  unsigned long long ga = (unsigned long long)(uintptr_t)g + (unsigned)(tid * 16);
  unsigned la = ldsB + (unsigned)(tid * 16);
  asm volatile(
      "global_load_async_to_lds_b128 %0, %1, off\n\t"
      "global_load_async_to_lds_b128 %0, %1, off offset:2048\n\t"
      "global_load_async_to_lds_b128 %0, %1, off offset:4096\n\t"
      "global_load_async_to_lds_b128 %0, %1, off offset:6144\n\t"
      "global_load_async_to_lds_b128 %0, %1, off offset:8192\n\t"
      "global_load_async_to_lds_b128 %0, %1, off offset:10240\n\t"
      "global_load_async_to_lds_b128 %0, %1, off offset:12288\n\t"
      "global_load_async_to_lds_b128 %0, %1, off offset:14336\n\t"
      "global_load_async_to_lds_b128 %0, %1, off offset:16384\n\t"
      "global_load_async_to_lds_b128 %0, %1, off offset:18432\n\t"
      "global_load_async_to_lds_b128 %0, %1, off offset:20480\n\t"
      "global_load_async_to_lds_b128 %0, %1, off offset:22528\n\t"
      "global_load_async_to_lds_b128 %0, %1, off offset:24576\n\t"
      "global_load_async_to_lds_b128 %0, %1, off offset:26624\n\t"
      "global_load_async_to_lds_b128 %0, %1, off offset:28672\n\t"
      "global_load_async_to_lds_b128 %0, %1, off offset:30720"
      :: "v"(la), "v"(ga) : "memory");
}

// ---------------------------------------------------------------------------
// Kernel 1: h_prime = h @ w per head, WMMA f32 16x16x4.
// w[hd] held in LDS in pair-interleaved layout so each B fragment is one
// aligned ds_load_b64 (no register shuffles). Emits hp pair-interleaved via
// b64 stores (rows r,r+1 are adjacent VGPRs of the D fragment).
// ---------------------------------------------------------------------------
__global__ __launch_bounds__(128) void hp_kernel(const float* __restrict__ h,
                                                 const float* __restrict__ w,
                                                 float* __restrict__ hp) {
  __shared__ float wshI[(FIN / 2) * 128];   // [k/2][col][k&1], 8 KB
  const int bh   = blockIdx.x;
  const int b    = bh / NH;
  const int hd   = bh % NH;
  const int tid  = threadIdx.x;
  const int wave = tid >> 5;
  const int lane = tid & 31;
  const int m    = lane & 15;
  const int half = lane >> 4;

  {  // stage w[hd] (32x64) interleaving row pairs: [p][c][0..1]
    const float4* src = (const float4*)(w + (size_t)hd * FIN * FOUT);
    #pragma unroll
    for (int it = 0; it < 2; ++it) {
      const int u  = tid + it * 128;        // 0..255 = 16 pairs x 16 col4
      const int p  = u >> 4;
      const int c4 = u & 15;
      const float4 r0 = src[(2 * p + 0) * 16 + c4];
      const float4 r1 = src[(2 * p + 1) * 16 + c4];
      float* dst = &wshI[p * 128 + c4 * 8];
      *(float2*)(dst + 0) = make_float2(r0.x, r1.x);
      *(float2*)(dst + 2) = make_float2(r0.y, r1.y);
      *(float2*)(dst + 4) = make_float2(r0.z, r1.z);
      *(float2*)(dst + 6) = make_float2(r0.w, r1.w);
    }
  }
  __syncthreads();

  const int n0 = blockIdx.y * 64 + wave * 16;
  v8f acc[4] = {};

  const float* hrow = h + ((size_t)b * NN + n0 + m) * FIN + 2 * half;
  for (int k = 0; k < FIN; k += 4) {
    v2f a;
    a.x = hrow[k + 0];
    a.y = hrow[k + 1];
    const int pr = ((k >> 1) + half) << 7;  // pair base in floats
    #pragma unroll
    for (int ot = 0; ot < 4; ++ot) {
      const v2f bb = *(const v2f*)(wshI + pr + ((ot * 16 + m) << 1));
      acc[ot] = __builtin_amdgcn_wmma_f32_16x16x4_f32(false, a, false, bb,
                                                      (short)0, acc[ot],
                                                      false, false);
    }
  }

  // interleaved store: rows (r, r+1) -> one float2 at [pair][col][0..1]
  float* dst = hp + (size_t)bh * NN * FOUT + (size_t)(n0 >> 1) * (2 * FOUT);
  #pragma unroll
  for (int ot = 0; ot < 4; ++ot) {
    const int col2 = (ot * 16 + m) * 2;
    #pragma unroll
    for (int r = 0; r < 8; r += 2) {
      const int j = r + 8 * half;  // even
      float2 o2 = make_float2(acc[ot][r], acc[ot][r + 1]);
      *(float2*)(dst + (size_t)(j >> 1) * 128 + col2) = o2;
    }
  }
}

// ---------------------------------------------------------------------------
// Kernel 2: s,d row dots against a_src/a_dst (reads interleaved hp)
// ---------------------------------------------------------------------------
__global__ __launch_bounds__(256) void sd_kernel(const float* __restrict__ hp,
                                                 const float* __restrict__ a_src,
                                                 const float* __restrict__ a_dst,
                                                 float* __restrict__ sarr,
                                                 float* __restrict__ darr) {
  const int t  = blockIdx.x * 256 + threadIdx.x;  // 0 .. BSZ*NH*NN-1
  const int n  = t & (NN - 1);
  const int bh = t >> 10;
  const int hd = bh & (NH - 1);
  const int q  = n & 1;
  const float4* P  = (const float4*)(hp + (size_t)bh * NN * FOUT + (size_t)(n >> 1) * 128);
  const float2* as = (const float2*)(a_src + hd * FOUT);
  const float2* ad = (const float2*)(a_dst + hd * FOUT);
  float s = 0.f, d = 0.f;
  #pragma unroll
  for (int u = 0; u < 32; ++u) {    // float4 = cols (2u,2u+1) x parities (0,1)
    const float4 v = P[u];
    const float e0 = q ? v.y : v.x;
    const float e1 = q ? v.w : v.z;
    const float2 sa = as[u];
    const float2 da = ad[u];
    s += e0 * sa.x + e1 * sa.y;
    d += e0 * da.x + e1 * da.y;
  }
  sarr[t] = s;
  darr[t] = d;
}

// ---------------------------------------------------------------------------
// Kernel 3: per (b,h): dmax; jpack[j]=(d',e^{d'},e^{0.2d'},0);
// ipack[i]=(t_i, A_i, B_i, 1/sum_i) with exact factorized softmax denominator.
// ---------------------------------------------------------------------------
__global__ __launch_bounds__(1024) void prep_kernel(const float* __restrict__ sarr,
                                                    const float* __restrict__ darr,
                                                    float4* __restrict__ jpack,
                                                    float4* __restrict__ ipack) {
  __shared__ float red[NN];
  __shared__ float shd[NN];
  __shared__ float shE[NN];
  __shared__ float shF[NN];
  const int bh = blockIdx.x;
  const int t  = threadIdx.x;

  const float dv = darr[bh * NN + t];
  red[t] = dv;
  __syncthreads();
  for (int s = NN / 2; s > 0; s >>= 1) {
    if (t < s) red[t] = fmaxf(red[t], red[t + s]);
    __syncthreads();
  }
  const float dmax = red[0];

  const float dsh = dv - dmax;
  const float E = expf(dsh);
  const float F = expf(NEG * dsh);
  shd[t] = dsh; shE[t] = E; shF[t] = F;
  jpack[bh * NN + t] = make_float4(dsh, E, F, 0.f);
  __syncthreads();

  const float sv = sarr[bh * NN + t];
  const float tt = sv + dmax;
  const float mm = (tt >= 0.f) ? tt : NEG * tt;
  const float Ai = expf(tt - mm);
  const float Bi = expf(NEG * tt - mm);
  float sum = 0.f;
  for (int j = 0; j < NN; ++j)
    sum += (shd[j] >= -tt) ? Ai * shE[j] : Bi * shF[j];
  ipack[bh * NN + t] = make_float4(tt, Ai, Bi, 1.0f / sum);
}

// ---------------------------------------------------------------------------
// Kernel 4: fused P + attn-write + (P @ h_prime), WMMA f32 16x16x4.
// Double-buffered async LDS staging (ASYNCcnt), B fragments as single b64.
// ---------------------------------------------------------------------------
__global__ __launch_bounds__(128) void attn_kernel(const float* __restrict__ hp,
                                                   const float4* __restrict__ jpack,
                                                   const float4* __restrict__ ipack,
                                                   const float* __restrict__ bias,
                                                   float* __restrict__ out,
                                                   float* __restrict__ attn) {
  __shared__ float  hpS[2][64 * 128];   // 2 x 32 KB, pair-interleaved chunks
  __shared__ float4 jpS[2][128];        // 2 x 2 KB
  const int bh   = blockIdx.x;
  const int tid  = threadIdx.x;
  const int wave = tid >> 5;
  const int lane = tid & 31;
  const int m    = lane & 15;
  const int half = lane >> 4;
  const int i    = blockIdx.y * 64 + wave * 16 + m;

  const float4 ip = ipack[bh * NN + i];
  const float tt = ip.x, Ai = ip.y, Bi = ip.z, inv = ip.w;
  float bv[4];
  #pragma unroll
  for (int ot = 0; ot < 4; ++ot) bv[ot] = bias[ot * 16 + m];

  const float* hpB  = hp + (size_t)bh * NN * FOUT;
  float*       arow = attn + ((size_t)bh * NN + i) * NN;

  const unsigned ldsB0 = ldsOffset(&hpS[0][0]);
  const unsigned ldsB1 = ldsOffset(&hpS[1][0]);

  v8f acc[4] = {};

  // prologue: stage chunk 0
  async_stage32k(hpB, ldsB0, tid);
  jpS[0][tid] = jpack[bh * NN + tid];

  for (int c = 0; c < NN / 128; ++c) {
    const int cb = c & 1;
    if (c + 1 < NN / 128) {   // overlap: issue next chunk before waiting
      async_stage32k(hpB + (size_t)(c + 1) * (128 * FOUT),
                     ((c + 1) & 1) ? ldsB1 : ldsB0, tid);
      jpS[(c + 1) & 1][tid] = jpack[bh * NN + (c + 1) * 128 + tid];
      asm volatile("s_wait_asynccnt 0x10" ::: "memory");  // chunk c complete
    } else {
      asm volatile("s_wait_asynccnt 0x0" ::: "memory");
    }
    __syncthreads();

    const float*  B = &hpS[cb][0];
    const float4* J = &jpS[cb][0];
    for (int jj = 0; jj < 128; jj += 4) {
      const int ja = jj + 2 * half;            // A cols / hp rows: ja, ja+1
      const float4 jp0 = J[ja];
      const float4 jp1 = J[ja + 1];
      float p0 = ((jp0.x >= -tt) ? Ai * jp0.y : Bi * jp0.z) * inv;
      float p1 = ((jp1.x >= -tt) ? Ai * jp1.y : Bi * jp1.z) * inv;
      v2f pv; pv.x = p0; pv.y = p1;
      __builtin_nontemporal_store(pv, (v2f*)(arow + c * 128 + ja));
      const int pr = ((jj >> 1) + half) << 7;  // pair base in floats
      #pragma unroll
      for (int ot = 0; ot < 4; ++ot) {
        const v2f bb = *(const v2f*)(B + pr + ((ot * 16 + m) << 1));
        acc[ot] = __builtin_amdgcn_wmma_f32_16x16x4_f32(false, pv, false, bb,
                                                        (short)0, acc[ot],
                                                        false, false);
      }
    }
    __syncthreads();
  }

  float* orow = out + ((size_t)bh * NN + blockIdx.y * 64 + wave * 16) * FOUT;
  #pragma unroll
  for (int ot = 0; ot < 4; ++ot)
    #pragma unroll
    for (int r = 0; r < 8; ++r)
      __builtin_nontemporal_store(acc[ot][r] + bv[ot],
          orow + (size_t)(r + 8 * half) * FOUT + ot * 16 + m);
}

// ---------------------------------------------------------------------------
extern "C" void kernel_launch(void* const* d_in, const int* in_sizes, int n_in,
                              void* d_out, int out_size, void* d_ws, size_t ws_size,
                              hipStream_t stream) {
  const float* h     = (const float*)d_in[0];
  const float* w     = (const float*)d_in[1];
  const float* a_src = (const float*)d_in[2];
  const float* a_dst = (const float*)d_in[3];
  const float* bias  = (const float*)d_in[4];

  float* out  = (float*)d_out;
  float* attn = out + (size_t)BSZ * NH * NN * FOUT;   // out first, then attn

  // workspace (floats): hp(interleaved) | sarr | darr | jpack(f4) | ipack(f4)
  float*  ws    = (float*)d_ws;
  float*  hp    = ws;                                  // 4,194,304
  float*  sarr  = hp + (size_t)BSZ * NH * NN * FOUT;   // 65,536
  float*  darr  = sarr + (size_t)BSZ * NH * NN;        // 65,536
  float4* jpack = (float4*)(darr + (size_t)BSZ * NH * NN);
  float4* ipack = jpack + (size_t)BSZ * NH * NN;       // total ~18.5 MB

  hp_kernel  <<<dim3(BSZ * NH, NN / 64), 128, 0, stream>>>(h, w, hp);
  sd_kernel  <<<(BSZ * NH * NN) / 256, 256, 0, stream>>>(hp, a_src, a_dst, sarr, darr);
  prep_kernel<<<BSZ * NH, NN, 0, stream>>>(sarr, darr, jpack, ipack);
  attn_kernel<<<dim3(BSZ * NH, NN / 64), 128, 0, stream>>>(hp, jpack, ipack, bias, out, attn);
}